// TopKPooler_85890755985655
// MI455X (gfx1250) — compile-verified
//
#include <hip/hip_runtime.h>
#include <cstddef>

typedef float v2f __attribute__((ext_vector_type(2)));
typedef float v8f __attribute__((ext_vector_type(8)));

namespace {
constexpr int Hn   = 200;
constexpr int Cn   = 100;
constexpr int Dn   = 128;
constexpr int HT   = (Hn + 15) / 16;   // 13 h-tiles
constexpr int HPAD = HT * 16;          // 208
constexpr int CSTR = Dn + 4;           // 132: LDS stride, kills bank conflicts
constexpr float kEps = 1e-12f;
}

__global__ __launch_bounds__(256) void topk_pool_wmma(
    const float* __restrict__ hist, const int* __restrict__ mask,
    const float* __restrict__ cand, float* __restrict__ out_score,
    float* __restrict__ out_emb)
{
  __shared__ float s_scores[16 * HPAD];   // [c 0..15][h 0..207]
  __shared__ float s_cand[16 * CSTR];     // normalized cand tile
  __shared__ float s_invH[HPAD];
  __shared__ int   s_mask[HPAD];
  __shared__ int   s_topidx[16][4];
  __shared__ float s_invden[16];
  __shared__ int   s_cnt[16];

  const int b    = blockIdx.y;
  const int c0   = blockIdx.x * 16;
  const int tid  = threadIdx.x;
  const int wave = tid >> 5;
  const int lane = tid & 31;
  const int half = lane >> 4;            // 0 or 1 (upper 16 lanes)
  const int l16  = lane & 15;

  const float* __restrict__ histB = hist + (size_t)b * Hn * Dn;
  const float* __restrict__ candB = cand + (size_t)b * Cn * Dn;

  // ---- stage L2-normalized candidate tile (16 x 128) into LDS ----
  #pragma unroll
  for (int rr = 0; rr < 2; ++rr) {
    const int r = wave * 2 + rr;
    const int c = min(c0 + r, Cn - 1);          // clamp; garbage rows never read back
    float4 v = *(const float4*)(candB + (size_t)c * Dn + lane * 4);
    float s = v.x * v.x + v.y * v.y + v.z * v.z + v.w * v.w;
    #pragma unroll
    for (int m = 16; m >= 1; m >>= 1) s += __shfl_xor(s, m, 32);
    const float inv = 1.0f / fmaxf(sqrtf(s), kEps);
    float* dst = &s_cand[r * CSTR + lane * 4];
    dst[0] = v.x * inv; dst[1] = v.y * inv; dst[2] = v.z * inv; dst[3] = v.w * inv;
  }

  // ---- inverse L2 norms of all history rows of this batch ----
  for (int r = wave; r < Hn; r += 8) {
    float4 v = *(const float4*)(histB + (size_t)r * Dn + lane * 4);
    float s = v.x * v.x + v.y * v.y + v.z * v.z + v.w * v.w;
    #pragma unroll
    for (int m = 16; m >= 1; m >>= 1) s += __shfl_xor(s, m, 32);
    if (lane == 0) s_invH[r] = 1.0f / fmaxf(sqrtf(s), kEps);
  }
  if (tid < HPAD) s_mask[tid] = (tid < Hn) ? mask[(size_t)b * Hn + tid] : 0;
  __syncthreads();

  // ---- scores[16 x 208] = candTile(16xD) * hist^T(D x H) via fp32 WMMA ----
  // A frag (16x4): lane -> M=l16, VGPR pair = K (+2 on upper half-lanes)
  // B frag (4x16): lane -> N=l16 (hist row h0+l16), same K split.
  // hist inv-norm is lane-constant over the output column, so it is applied
  // ONCE to the accumulator (not per K-step) -> WMMA consumes loads directly.
  const float* ap = &s_cand[l16 * CSTR + half * 2];
  for (int nt = wave; nt < HT; nt += 8) {          // wave-uniform: EXEC all 1s
    const int h0   = nt * 16;
    const int hrow = min(h0 + l16, Hn - 1);        // branchless clamp
    const float scale = s_invH[hrow];              // per-column (lane-constant)
    const float* bp = histB + (size_t)hrow * Dn + half * 2;
    {   // prefetch this wave's next h-tile (global_prefetch_b8)
      const int hnext = min(h0 + 8 * 16 + l16, Hn - 1);
      __builtin_prefetch(histB + (size_t)hnext * Dn + half * 2, 0, 3);
    }
    v8f acc = {};
    #pragma unroll
    for (int k0 = 0; k0 < Dn; k0 += 4) {
      v2f a  = *(const v2f*)(ap + k0);             // ds_load_b64 (hoisted/resident)
      v2f bb = *(const v2f*)(bp + k0);             // global_load_b64, raw
      acc = __builtin_amdgcn_wmma_f32_16x16x4_f32(
          false, a, false, bb, (short)0, acc, false, false);
    }
    // D layout: VGPR j -> row M=j (+8 upper lanes), col N = l16
    const int col = h0 + l16;
    const int r0  = half * 8;
    #pragma unroll
    for (int j = 0; j < 8; ++j)
      s_scores[(r0 + j) * HPAD + col] = acc[j] * scale;   // fold invH here
  }
  __syncthreads();

  // ---- per-candidate masked top-4 (sorted insertion; first-index tiebreak) ----
  if (tid < 16) {
    const float* row = &s_scores[tid * HPAD];
    float v0 = -1e30f, v1 = -1e30f, v2 = -1e30f, v3 = -1e30f;
    int   i0 = -1, i1 = -1, i2 = -1, i3 = -1;
    for (int h = 0; h < Hn; ++h) {
      if (s_mask[h] == 0) continue;                // masked == never a top-k "valid"
      const float s = row[h];
      if (s > v3) {
        if (s > v0)      { v3=v2; i3=i2; v2=v1; i2=i1; v1=v0; i1=i0; v0=s; i0=h; }
        else if (s > v1) { v3=v2; i3=i2; v2=v1; i2=i1; v1=s;  i1=h; }
        else if (s > v2) { v3=v2; i3=i2; v2=s;  i2=h; }
        else             { v3=s;  i3=h; }
      }
    }
    const int   cnt   = (i0 >= 0) + (i1 >= 0) + (i2 >= 0) + (i3 >= 0);
    const float denom = fmaxf((float)cnt, 1.0f);
    const float ssum  = (i0 >= 0 ? v0 : 0.f) + (i1 >= 0 ? v1 : 0.f) +
                        (i2 >= 0 ? v2 : 0.f) + (i3 >= 0 ? v3 : 0.f);
    s_topidx[tid][0] = i0; s_topidx[tid][1] = i1;
    s_topidx[tid][2] = i2; s_topidx[tid][3] = i3;
    s_cnt[tid]    = cnt;
    s_invden[tid] = 1.0f / denom;
    const int c = c0 + tid;
    if (c < Cn) out_score[(size_t)b * Cn + c] = ssum / denom;
  }
  __syncthreads();

  // ---- mean of gathered normalized embeddings: coalesced over D ----
  {
    const int dg = tid & 127;                      // d index
    for (int cc = tid >> 7; cc < 16; cc += 2) {    // two 128-thread groups
      const int c = c0 + cc;
      if (c >= Cn) break;
      const int cnt = s_cnt[cc];                   // uniform within group
      float sum = 0.0f;
      #pragma unroll
      for (int j = 0; j < 4; ++j) {
        if (j < cnt) {
          const int h = s_topidx[cc][j];
          sum += histB[(size_t)h * Dn + dg] * s_invH[h];
        }
      }
      out_emb[((size_t)b * Cn + c) * Dn + dg] = sum * s_invden[cc];
    }
  }
}

extern "C" void kernel_launch(void* const* d_in, const int* in_sizes, int n_in,
                              void* d_out, int out_size, void* d_ws, size_t ws_size,
                              hipStream_t stream) {
  (void)n_in; (void)out_size; (void)d_ws; (void)ws_size;
  const float* hist = (const float*)d_in[0];   // [B,H,D] f32
  const int*   mask = (const int*)d_in[1];     // [B,H] i32
  const float* cand = (const float*)d_in[2];   // [B,C,D] f32
  const int B = in_sizes[1] / Hn;              // in_sizes[1] == B*H
  float* out_score = (float*)d_out;                    // [B,C]
  float* out_emb   = out_score + (size_t)B * Cn;       // [B,C,D]
  dim3 grid((Cn + 15) / 16, B);
  topk_pool_wmma<<<grid, 256, 0, stream>>>(hist, mask, cand, out_score, out_emb);
}